// dist_loss_62929860821443
// MI455X (gfx1250) — compile-verified
//
#include <hip/hip_runtime.h>
#include <hip/hip_bf16.h>

typedef float v2f __attribute__((ext_vector_type(2)));
typedef float v8f __attribute__((ext_vector_type(8)));

// ---------------------------------------------------------------------------
// Per-wave (wave32) sum reduction.
// CDNA5 path: one V_WMMA_F32_16X16X4_F32 with A = lane accumulators, B = ones.
//   A is 16x4 F32: lanes 0-15 hold A[m][0] (VGPR0) / A[m][1] (VGPR1),
//                  lanes 16-31 hold A[m][2] / A[m][3].
//   We place acc in the first A VGPR and 0 in the second, B = all 1.0, so
//   D[m][n] = acc_m + acc_{m+16} for every column n.
//   D layout: VGPR v, lanes 0-15 -> row M=v, lanes 16-31 -> row M=v+8.
//   Summing d[0..7] per lane yields rows 0-7 total (lanes 0-15) and rows 8-15
//   total (lanes 16-31); one xor-16 shuffle gives the full 32-lane sum.
// ---------------------------------------------------------------------------
__device__ __forceinline__ float wave_reduce_sum(float acc) {
#if defined(__gfx1250__) && __has_builtin(__builtin_amdgcn_wmma_f32_16x16x4_f32)
    v2f a;
    a.x = acc;
    a.y = 0.0f;
    v2f b;
    b.x = 1.0f;
    b.y = 1.0f;
    v8f c = {};
    c = __builtin_amdgcn_wmma_f32_16x16x4_f32(
        /*neg_a=*/false, a, /*neg_b=*/false, b,
        /*c_mod=*/(short)0, c, /*reuse_a=*/false, /*reuse_b=*/false);
    float s = ((c[0] + c[1]) + (c[2] + c[3])) + ((c[4] + c[5]) + (c[6] + c[7]));
    s += __shfl_xor(s, 16, 32);
    return s;
#else
    #pragma unroll
    for (int off = 16; off > 0; off >>= 1) acc += __shfl_xor(acc, off, 32);
    return acc;
#endif
}

// ---------------------------------------------------------------------------
// Kernel 1: streaming signed reduction. S = sum((target==0) ? pred : -pred).
// 128-bit vector loads: float4 for pred, longlong4 for target (2x b128).
// One partial per block (deterministic tree), written to d_ws.
// ---------------------------------------------------------------------------
__global__ void dist_loss_reduce_kernel(const float* __restrict__ pred,
                                        const long long* __restrict__ target,
                                        float* __restrict__ partials,
                                        int n) {
    const int tid      = blockIdx.x * blockDim.x + threadIdx.x;
    const int nthreads = gridDim.x * blockDim.x;

    float acc = 0.0f;

    const int n4 = n >> 2;  // groups of 4 elements
    const float4*    p4 = (const float4*)pred;
    const longlong4* t4 = (const longlong4*)target;

    for (int i = tid; i < n4; i += nthreads) {
        float4    p = p4[i];
        longlong4 t = t4[i];
        acc += (t.x == 0) ? p.x : -p.x;
        acc += (t.y == 0) ? p.y : -p.y;
        acc += (t.z == 0) ? p.z : -p.z;
        acc += (t.w == 0) ? p.w : -p.w;
    }

    // Tail (n not a multiple of 4) — converges before the WMMA below,
    // so EXEC is all-ones again when the matrix op issues.
    for (int i = (n4 << 2) + tid; i < n; i += nthreads) {
        float p = pred[i];
        acc += (target[i] == 0) ? p : -p;
    }

    // Full-EXEC wave reduction (emits v_wmma_f32_16x16x4_f32).
    float wsum = wave_reduce_sum(acc);

    __shared__ float sm[8];  // 256 threads = 8 wave32s
    const int wave = threadIdx.x >> 5;
    const int lane = threadIdx.x & 31;
    if (lane == 0) sm[wave] = wsum;
    __syncthreads();

    if (threadIdx.x == 0) {
        float b = 0.0f;
        #pragma unroll
        for (int j = 0; j < 8; ++j) b += sm[j];
        partials[blockIdx.x] = b;
    }
}

// ---------------------------------------------------------------------------
// Kernel 2: fold block partials, apply loss epilogue.
// loss = max(0, 1 + S * (1/t)),  t = n/2.
// ---------------------------------------------------------------------------
__global__ void dist_loss_finalize_kernel(const float* __restrict__ partials,
                                          int np,
                                          float* __restrict__ out,
                                          float inv_t) {
    __shared__ float sm[256];
    float s = 0.0f;
    for (int i = threadIdx.x; i < np; i += 256) s += partials[i];
    sm[threadIdx.x] = s;
    __syncthreads();
    #pragma unroll
    for (int off = 128; off > 0; off >>= 1) {
        if (threadIdx.x < off) sm[threadIdx.x] += sm[threadIdx.x + off];
        __syncthreads();
    }
    if (threadIdx.x == 0) {
        float loss = 1.0f + sm[0] * inv_t;
        out[0] = (loss > 0.0f) ? loss : 0.0f;
    }
}

extern "C" void kernel_launch(void* const* d_in, const int* in_sizes, int n_in,
                              void* d_out, int out_size, void* d_ws, size_t ws_size,
                              hipStream_t stream) {
    const float*     pred   = (const float*)d_in[0];
    const long long* target = (const long long*)d_in[1];
    float*           out    = (float*)d_out;
    float*           parts  = (float*)d_ws;

    const int n = in_sizes[0];           // 8,388,608
    const int blocks  = 1024;            // 1024 x 256 = 8192 wave32s in flight
    const int threads = 256;

    const float inv_t = 2.0f / (float)n; // 1 / (n/2)

    dist_loss_reduce_kernel<<<blocks, threads, 0, stream>>>(pred, target, parts, n);
    dist_loss_finalize_kernel<<<1, 256, 0, stream>>>(parts, blocks, out, inv_t);
}